// KAST_49692771615002
// MI455X (gfx1250) — compile-verified
//
#include <hip/hip_runtime.h>
#include <hip/hip_bf16.h>

typedef __attribute__((ext_vector_type(2))) float v2f;
typedef __attribute__((ext_vector_type(8))) float v8f;

#define BS       2
#define MM       100
#define NN       4096
#define CK       256
#define CV       3
#define PP       225
#define CENTER_P 113
#define MT       7        // ceil(100/16) m-tiles

// ---------------------------------------------------------------------------
// Kernel 1: inner[b,n,m] = k_next[b,n,:] . m_k[b,m,CENTER,:]  via fp32 WMMA,
// then argmax over m. One wave32 per 16-query tile.
// D = A(16xK centers) x B(Kx16 queries): lane = query column, VGPRs = m rows.
// ---------------------------------------------------------------------------
__global__ __launch_bounds__(32)
void KAST_argmax_wmma(const float* __restrict__ m_k,
                      const float* __restrict__ k_next,
                      int* __restrict__ best_idx) {
    const int b    = blockIdx.y;
    const int n0   = blockIdx.x * 16;
    const int lane = threadIdx.x;      // wave32
    const int col  = lane & 15;        // query column / center row held by lane
    const int kh   = lane >> 4;        // K half selector (per 16x16x4 layouts)

    // A operand row pointers: center rows m = t*16 + col (clamped in-bounds,
    // masked out of the argmax below).
    const float* aptr[MT];
#pragma unroll
    for (int t = 0; t < MT; ++t) {
        int m = t * 16 + col;
        if (m > MM - 1) m = MM - 1;
        aptr[t] = m_k + (((size_t)b * MM + m) * PP + CENTER_P) * CK;
    }
    const float* bptr = k_next + ((size_t)b * NN + (n0 + col)) * CK;

    v8f acc[MT] = {};   // 7 x (16x16 f32) accumulator tiles

    for (int kb = 0; kb < CK; kb += 4) {
        const int ko = kb + 2 * kh;            // lane's K pair within this step
        float2 bv = *(const float2*)(bptr + ko);
        v2f bb; bb[0] = bv.x; bb[1] = bv.y;
#pragma unroll
        for (int t = 0; t < MT; ++t) {
            float2 av = *(const float2*)(aptr[t] + ko);
            v2f aa; aa[0] = av.x; aa[1] = av.y;
            acc[t] = __builtin_amdgcn_wmma_f32_16x16x4_f32(
                false, aa, false, bb, (short)0, acc[t], false, false);
        }
    }

    // In-lane argmax over this lane's 56 m rows (m = t*16 + v + 8*kh).
    float bestv = -3.0e38f;
    int   bestm = 0;
#pragma unroll
    for (int t = 0; t < MT; ++t) {
#pragma unroll
        for (int v = 0; v < 8; ++v) {
            const int m = t * 16 + v + 8 * kh;
            const float val = acc[t][v];
            if (m < MM && val > bestv) { bestv = val; bestm = m; }
        }
    }
    // Merge the two half-wave row blocks holding the same query column.
    const float ov = __shfl_xor(bestv, 16, 32);
    const int   om = __shfl_xor(bestm, 16, 32);
    if (ov > bestv || (ov == bestv && om < bestm)) { bestv = ov; bestm = om; }

    if (lane < 16) best_idx[b * NN + n0 + col] = bestm;
}

// ---------------------------------------------------------------------------
// Kernel 2: gather winning patch, logits over 225 taps, softmax, 3-channel
// weighted sum. One 256-thread block per (b,n); m_k patches live in L2.
// ---------------------------------------------------------------------------
__global__ __launch_bounds__(256)
void KAST_attend(const float* __restrict__ m_k,
                 const float* __restrict__ k_next,
                 const float* __restrict__ m_v,
                 const int*   __restrict__ best_idx,
                 float* __restrict__ out) {
    __shared__ float q[CK];
    __shared__ float red[256];

    const int b = blockIdx.y;
    const int n = blockIdx.x;
    const int t = threadIdx.x;
    const int best = best_idx[b * NN + n];

    // Stage query vector in LDS (broadcast reads below).
    q[t] = k_next[((size_t)b * NN + n) * CK + t];
    __syncthreads();

    float logit = -1.0e30f;
    float mv0 = 0.f, mv1 = 0.f, mv2 = 0.f;
    if (t < PP) {
        const float* row = m_k + (((size_t)b * MM + best) * PP + t) * CK;
        float acc = 0.f;
#pragma unroll 4
        for (int c = 0; c < CK; c += 4) {
            float4 r = *(const float4*)(row + c);
            acc += r.x * q[c] + r.y * q[c + 1] + r.z * q[c + 2] + r.w * q[c + 3];
        }
        logit = acc;
        const float* vrow = m_v + (((size_t)b * MM + best) * PP + t) * CV;
        mv0 = vrow[0]; mv1 = vrow[1]; mv2 = vrow[2];
    }

    // Block max
    red[t] = logit; __syncthreads();
    for (int s = 128; s > 0; s >>= 1) {
        if (t < s) { float o = red[t + s]; if (o > red[t]) red[t] = o; }
        __syncthreads();
    }
    const float mx = red[0]; __syncthreads();

    // exp + block sum
    float w = (t < PP) ? __expf(logit - mx) : 0.f;
    red[t] = w; __syncthreads();
    for (int s = 128; s > 0; s >>= 1) {
        if (t < s) red[t] += red[t + s];
        __syncthreads();
    }
    const float inv = 1.0f / red[0]; __syncthreads();
    w *= inv;

    // 3 channel reductions of w * m_v
    float ch0 = w * mv0, ch1 = w * mv1, ch2 = w * mv2;
    float chs[3] = {ch0, ch1, ch2};
#pragma unroll
    for (int c = 0; c < CV; ++c) {
        red[t] = chs[c]; __syncthreads();
        for (int s = 128; s > 0; s >>= 1) {
            if (t < s) red[t] += red[t + s];
            __syncthreads();
        }
        if (t == 0) out[((size_t)b * NN + n) * CV + c] = red[0];
        __syncthreads();
    }
}

// ---------------------------------------------------------------------------
extern "C" void kernel_launch(void* const* d_in, const int* in_sizes, int n_in,
                              void* d_out, int out_size, void* d_ws, size_t ws_size,
                              hipStream_t stream) {
    const float* m_k    = (const float*)d_in[0];
    const float* k_next = (const float*)d_in[1];
    const float* m_v    = (const float*)d_in[2];
    float* out = (float*)d_out;
    int*   best = (int*)d_ws;   // BS*NN ints = 32 KB of scratch

    dim3 g1(NN / 16, BS);       // 512 waves, one wave per 16-query tile
    KAST_argmax_wmma<<<g1, 32, 0, stream>>>(m_k, k_next, best);

    dim3 g2(NN, BS);            // one block per (b, n)
    KAST_attend<<<g2, 256, 0, stream>>>(m_k, k_next, m_v, best, out);
}